// S4DKernel_75728863363112
// MI455X (gfx1250) — compile-verified
//
#include <hip/hip_runtime.h>
#include <math.h>

// ---------------------------------------------------------------------------
// S4D kernel generation via two-level Vandermonde decomposition + f32 WMMA.
//   l = 64*q + r  =>  K[n,l] = P[n,q] * Q[n,r]
//   out[h] (64x64 in q,r) = G (64 x 64) @ Qm (64 x 64), k = 2*N2 (re/im split)
// One workgroup (4 wave32s) per head h.
// ---------------------------------------------------------------------------

typedef __attribute__((ext_vector_type(2))) float v2f;
typedef __attribute__((ext_vector_type(8))) float v8f;

#define H_DIM  1024
#define N2_DIM 32
#define L_DIM  4096
#define QD     64     // # q values  (L / 64)
#define RD     64     // # r values
#define KD     64     // reduction dim = 2*N2
#define GS     66     // G LDS row stride (padded: 2*q mod 64 -> conflict-free A reads)
#define QS     65     // Qm LDS row stride (padded: kills phase-1 write conflicts)

__global__ __launch_bounds__(128) void s4d_wmma_kernel(
    const float* __restrict__ log_dt,      // (H)
    const float* __restrict__ C,           // (H, N2, 2)
    const float* __restrict__ log_A_real,  // (H, N2)
    const float* __restrict__ A_imag,      // (H, N2)
    float* __restrict__ out)               // (H, L)
{
    __shared__ float Gs[QD * GS];   // G[q][k]   : k=2n -> Re(Cc'*P), k=2n+1 -> Im
    __shared__ float Qs[KD * QS];   // Qm[k][r]  : k=2n -> Re(Q),     k=2n+1 -> -Im(Q)

    const int h = blockIdx.x;
    const int t = threadIdx.x;          // 0..127
    const int n = t & (N2_DIM - 1);     // 0..31
    const int g = t >> 5;               // 0..3 (wave id; 4 threads share each n)

    // ---------------- per-(h,n) parameters (registers) ----------------
    const float dt  = expf(log_dt[h]);
    const float Are = -expf(log_A_real[h * N2_DIM + n]);
    const float Aim = A_imag[h * N2_DIM + n];
    const float dAr = Are * dt;
    const float dAi = Aim * dt;

    // Cc' = 2 * (C0 + i C1) * (exp(dtA) - 1) / A
    const float em = expf(dAr);
    float es, ec;
    sincosf(dAi, &es, &ec);
    const float numr = em * ec - 1.0f;
    const float numi = em * es;
    const float inv  = 1.0f / (Are * Are + Aim * Aim);
    const float fr   = (numr * Are + numi * Aim) * inv;
    const float fi   = (numi * Are - numr * Aim) * inv;
    const float C0   = C[(h * N2_DIM + n) * 2 + 0];
    const float C1   = C[(h * N2_DIM + n) * 2 + 1];
    const float ccr  = 2.0f * (C0 * fr - C1 * fi);
    const float cci  = 2.0f * (C0 * fi + C1 * fr);

    // ---------------- phase 1: build G and Qm in LDS ----------------
    // thread (n, g) fills q,r in [16g, 16g+16) for its n.
    #pragma unroll 4
    for (int j = 0; j < 16; ++j) {
        const int q = g * 16 + j;
        // P[n,q] = exp(dtA * 64q)
        const float lq = (float)(q << 6);
        const float pm = expf(dAr * lq);
        float ps, pc;
        sincosf(dAi * lq, &ps, &pc);
        const float Pr = pm * pc, Pi = pm * ps;
        Gs[q * GS + 2 * n]     = ccr * Pr - cci * Pi;   // Re(Cc' * P)
        Gs[q * GS + 2 * n + 1] = ccr * Pi + cci * Pr;   // Im(Cc' * P)

        const int r = q;
        const float lr = (float)r;
        const float qm = expf(dAr * lr);
        float qs, qc;
        sincosf(dAi * lr, &qs, &qc);
        Qs[(2 * n) * QS + r]     =  qm * qc;            //  Re(Q)
        Qs[(2 * n + 1) * QS + r] = -qm * qs;            // -Im(Q)
    }
    __syncthreads();

    // ---------------- phase 2: 64x64x64 GEMM via V_WMMA_F32_16X16X4_F32 ----
    // wave g owns output rows q in [16g, 16g+16); 4 r-tiles of 16.
    const int lane = t & 31;
    const int hi   = lane >> 4;         // lane half (selects K pair per ISA layout)
    const int lm   = lane & 15;
    const int qrow = g * 16 + lm;       // A-fragment row (M = lane mod 16)

    v8f acc0 = {}, acc1 = {}, acc2 = {}, acc3 = {};

    #pragma unroll
    for (int kk = 0; kk < 16; ++kk) {
        const int kb = kk * 4 + 2 * hi;
        // A 16x4 f32: VGPR0 = K{kb}, VGPR1 = K{kb+1} (per half-wave)
        v2f a;
        a.x = Gs[qrow * GS + kb];
        a.y = Gs[qrow * GS + kb + 1];
        // B 4x16 f32: row kb (x) / kb+1 (y), N = lane mod 16
        v2f b0, b1, b2, b3;
        b0.x = Qs[kb * QS +  0 + lm];  b0.y = Qs[(kb + 1) * QS +  0 + lm];
        b1.x = Qs[kb * QS + 16 + lm];  b1.y = Qs[(kb + 1) * QS + 16 + lm];
        b2.x = Qs[kb * QS + 32 + lm];  b2.y = Qs[(kb + 1) * QS + 32 + lm];
        b3.x = Qs[kb * QS + 48 + lm];  b3.y = Qs[(kb + 1) * QS + 48 + lm];
        // 4 independent accumulator chains -> WMMAs pipeline back-to-back
        acc0 = __builtin_amdgcn_wmma_f32_16x16x4_f32(false, a, false, b0, (short)0, acc0, false, false);
        acc1 = __builtin_amdgcn_wmma_f32_16x16x4_f32(false, a, false, b1, (short)0, acc1, false, false);
        acc2 = __builtin_amdgcn_wmma_f32_16x16x4_f32(false, a, false, b2, (short)0, acc2, false, false);
        acc3 = __builtin_amdgcn_wmma_f32_16x16x4_f32(false, a, false, b3, (short)0, acc3, false, false);
    }

    // ---------------- store: C/D layout VGPR i -> M = i + 8*hi, N = lm ------
    float* __restrict__ outh = out + (size_t)h * L_DIM;
    #pragma unroll
    for (int i = 0; i < 8; ++i) {
        const int q = g * 16 + i + 8 * hi;
        outh[q * RD +  0 + lm] = acc0[i];
        outh[q * RD + 16 + lm] = acc1[i];
        outh[q * RD + 32 + lm] = acc2[i];
        outh[q * RD + 48 + lm] = acc3[i];
    }
}

extern "C" void kernel_launch(void* const* d_in, const int* in_sizes, int n_in,
                              void* d_out, int out_size, void* d_ws, size_t ws_size,
                              hipStream_t stream) {
    // inputs (setup_inputs order): [0]=L (int scalar), [1]=log_dt (H),
    // [2]=C (H,N2,2), [3]=log_A_real (H,N2), [4]=A_imag (H,N2)
    const float* log_dt     = (const float*)d_in[1];
    const float* C          = (const float*)d_in[2];
    const float* log_A_real = (const float*)d_in[3];
    const float* A_imag     = (const float*)d_in[4];
    float* out = (float*)d_out;

    dim3 grid(H_DIM);
    dim3 block(128);   // 4 wave32s per workgroup, one workgroup per head
    hipLaunchKernelGGL(s4d_wmma_kernel, grid, block, 0, stream,
                       log_dt, C, log_A_real, A_imag, out);
}